// TopSampler_59485297050203
// MI455X (gfx1250) — compile-verified
//
#include <hip/hip_runtime.h>

// CDNA5 / gfx1250, wave32.
// TopSampler: per (b,h): scores = q[b,h,0,:] . k[b,h,1:,:]  (f32, exact via
// V_WMMA_F32_16X16X4_F32), then mark sorted positions whose source index is
// among the first 1024 scores. out[b,h,0]=1; out[b,h,rank(i)+1]=1 for i<1024.

typedef __attribute__((ext_vector_type(2))) float v2f;
typedef __attribute__((ext_vector_type(8))) float v8f;

#define S      4096      // sequence length
#define NSCORE 4095      // keys 1..4095
#define D      64        // head dim
#define RTHR   1024      // top-r threshold

__global__ __launch_bounds__(256)
void topsampler_kernel(const float* __restrict__ q,
                       const float* __restrict__ k,
                       float* __restrict__ out) {
    const int bh  = blockIdx.x;          // 0..127  (b*16+h)
    const int tid = threadIdx.x;         // 0..255
    const int wave = tid >> 5;           // 0..7
    const int lane = tid & 31;           // 0..31
    const int col  = lane & 15;          // N column / key-in-tile
    const int half = lane >> 4;          // 0: K=0,1   1: K=2,3

    __shared__ __align__(16) float s_q[D];
    __shared__ __align__(16) float s_scores[S];   // s_scores[i] = score of key i+1
    __shared__ __align__(16) float s_mask[S];

    const float* qb = q + (size_t)bh * S * D;     // row 0 of q for this (b,h)
    const float* kb = k + (size_t)bh * S * D;

    // ---- stage 0: q -> LDS, zero mask -------------------------------------
    if (tid < D) s_q[tid] = qb[tid];
#pragma unroll
    for (int r = 0; r < 16; ++r) s_mask[tid + 256 * r] = 0.0f;
    __syncthreads();

    // ---- stage 1: scores via V_WMMA_F32_16X16X4_F32 -----------------------
    // A (16x4) = q chunk broadcast to all 16 rows; B (4x16) = 16 keys.
    // A/B 32-bit layout: lane L(0-15): VGPR0=K0,VGPR1=K1 ; lane L+16: K2,K3.
    // D layout: lane n holds column N=n&15 (all rows identical) -> lanes 0-15
    // write one score each per tile.
    for (int m = 0; m < 32; ++m) {
        const int tile = wave * 32 + m;               // 0..255
        int key = tile * 16 + col + 1;                // 1..4096
        key = min(key, S - 1);                        // clamp OOB key 4096
        const float* krow = kb + (size_t)key * D;

        v8f acc = {0.f, 0.f, 0.f, 0.f, 0.f, 0.f, 0.f, 0.f};
#pragma unroll
        for (int k0 = 0; k0 < D; k0 += 4) {
            const int ko = k0 + half * 2;
            v2f av, bv;
            av.x = s_q[ko];
            av.y = s_q[ko + 1];
            bv.x = krow[ko];
            bv.y = krow[ko + 1];
            acc = __builtin_amdgcn_wmma_f32_16x16x4_f32(
                false, av, false, bv, (short)0, acc, false, false);
        }
        if (half == 0) {
            const int sidx = tile * 16 + col;         // 0..4095
            s_scores[sidx] = (sidx == NSCORE) ? __builtin_inff() : acc[0];
        }
    }
    __syncthreads();

    // ---- stage 2: rank first 1024 scores, scatter into LDS mask -----------
    float ai[4];
    int   ii[4];
    int   cnt[4] = {0, 0, 0, 0};
#pragma unroll
    for (int t = 0; t < 4; ++t) {
        ii[t] = tid + t * 256;                        // 0..1023
        ai[t] = s_scores[ii[t]];
    }
    const float4* sc4 = (const float4*)s_scores;
    for (int jj = 0; jj < S / 4; ++jj) {
        const float4 v = sc4[jj];                     // LDS broadcast, b128
        const float vc[4] = {v.x, v.y, v.z, v.w};
#pragma unroll
        for (int c = 0; c < 4; ++c) {
            const int j = jj * 4 + c;
#pragma unroll
            for (int t = 0; t < 4; ++t) {
                // stable ascending rank: smaller value, or equal & earlier idx
                cnt[t] += (vc[c] < ai[t]) || ((vc[c] == ai[t]) && (j < ii[t]));
            }
        }
    }
#pragma unroll
    for (int t = 0; t < 4; ++t) s_mask[cnt[t] + 1] = 1.0f;   // ranks distinct
    if (tid == 0) s_mask[0] = 1.0f;
    __syncthreads();

    // ---- stage 3: coalesced writeback -------------------------------------
    float4* out4 = (float4*)(out + (size_t)bh * S);
    const float4* m4 = (const float4*)s_mask;
#pragma unroll
    for (int r = 0; r < 4; ++r) {
        const int idx = r * 256 + tid;                // 0..1023 float4s
        out4[idx] = m4[idx];
    }
}

extern "C" void kernel_launch(void* const* d_in, const int* in_sizes, int n_in,
                              void* d_out, int out_size, void* d_ws, size_t ws_size,
                              hipStream_t stream) {
    const float* q = (const float*)d_in[0];   // (8,16,4096,64) f32
    const float* k = (const float*)d_in[1];   // (8,16,4096,64) f32
    float* out = (float*)d_out;               // (8,16,4096) mask as float
    (void)in_sizes; (void)n_in; (void)out_size; (void)d_ws; (void)ws_size;

    topsampler_kernel<<<dim3(128), dim3(256), 0, stream>>>(q, k, out);
}